// LightGCN_27444841021791
// MI455X (gfx1250) — compile-verified
//
#include <hip/hip_runtime.h>
#include <hip/hip_bf16.h>
#include <math.h>

// ---------------- problem constants (match reference) ----------------
#define N_USERS_C 100000
#define N_ITEMS_C 50000
#define NNODES    150000          // N_USERS + N_ITEMS
#define NNZ_E     4000000
#define DIM_C     64
#define NLAYERS   3
#define BATCH_C   4096
#define V4ROW     (DIM_C / 4)     // 16 float4 per embedding row

typedef __attribute__((ext_vector_type(2))) float v2f;
typedef __attribute__((ext_vector_type(8))) float v8f;

// =====================================================================
// init: x_cur = acc = concat(user_emb, item_emb)
// =====================================================================
__global__ void k_init(const float* __restrict__ ue, const float* __restrict__ ie,
                       float* __restrict__ x, float* __restrict__ acc) {
  int i = blockIdx.x * blockDim.x + threadIdx.x;          // float4 index
  const int total4 = NNODES * V4ROW;                      // 2.4M
  if (i >= total4) return;
  int node = i / V4ROW;
  int off  = i - node * V4ROW;
  const float4* src = (node < N_USERS_C)
      ? ((const float4*)ue) + (size_t)node * V4ROW + off
      : ((const float4*)ie) + (size_t)(node - N_USERS_C) * V4ROW + off;
  float4 v = *src;
  ((float4*)x)[i]   = v;
  ((float4*)acc)[i] = v;
}

__global__ void k_zero_u32(unsigned* __restrict__ p, int n) {
  int i = blockIdx.x * blockDim.x + threadIdx.x;
  if (i < n) p[i] = 0u;
}

__global__ void k_zero_f4(float4* __restrict__ p, int n4) {
  int i = blockIdx.x * blockDim.x + threadIdx.x;
  if (i < n4) { float4 z; z.x = z.y = z.z = z.w = 0.f; p[i] = z; }
}

// =====================================================================
// CSR build: histogram -> exclusive scan (3 kernels) -> fill
// =====================================================================
__global__ void k_hist(const int* __restrict__ edst, unsigned* __restrict__ counts) {
  int e = blockIdx.x * blockDim.x + threadIdx.x;
  if (e < NNZ_E) atomicAdd(&counts[edst[e]], 1u);
}

// per-block exclusive scan of 256 elements + block sums (LDS Hillis-Steele)
__global__ void k_scan1(const unsigned* __restrict__ in, unsigned* __restrict__ out,
                        unsigned* __restrict__ bsums, int n) {
  __shared__ unsigned s[256];
  int t = threadIdx.x;
  int g = blockIdx.x * 256 + t;
  unsigned v = (g < n) ? in[g] : 0u;
  s[t] = v;
  __syncthreads();
  for (int o = 1; o < 256; o <<= 1) {
    unsigned u = (t >= o) ? s[t - o] : 0u;
    __syncthreads();
    s[t] += u;
    __syncthreads();
  }
  unsigned incl = s[t];
  if (g < n) out[g] = incl - v;                 // exclusive
  if (t == 255) bsums[blockIdx.x] = incl;       // block total
}

// scan the (<=1024) block sums in place, single block
__global__ void k_scan2(unsigned* __restrict__ bsums, int nb) {
  __shared__ unsigned s[1024];
  int t = threadIdx.x;
  unsigned v = (t < nb) ? bsums[t] : 0u;
  s[t] = v;
  __syncthreads();
  for (int o = 1; o < 1024; o <<= 1) {
    unsigned u = (t >= o) ? s[t - o] : 0u;
    __syncthreads();
    s[t] += u;
    __syncthreads();
  }
  if (t < nb) bsums[t] = s[t] - v;              // exclusive block offsets
}

// add block offsets, copy row starts into fill cursors, close row_ptr
__global__ void k_scan3(unsigned* __restrict__ rowptr, const unsigned* __restrict__ bsums,
                        unsigned* __restrict__ cursor, int n) {
  int g = blockIdx.x * 256 + threadIdx.x;
  if (g < n) {
    unsigned v = rowptr[g] + bsums[blockIdx.x];
    rowptr[g] = v;
    cursor[g] = v;
  }
  if (g == 0) rowptr[n] = (unsigned)NNZ_E;
}

__global__ void k_fill(const int* __restrict__ esrc, const int* __restrict__ edst,
                       const float* __restrict__ eval, unsigned* __restrict__ cursor,
                       int* __restrict__ csrc, float* __restrict__ cval) {
  int e = blockIdx.x * blockDim.x + threadIdx.x;
  if (e >= NNZ_E) return;
  unsigned p = atomicAdd(&cursor[edst[e]], 1u);
  csrc[p] = esrc[e];
  cval[p] = eval[e];
}

// =====================================================================
// CSR SpMM: one 16-lane group owns one dst row (lane -> float4 of dims).
// No float atomics; acc update fused (row is written exactly once).
// Working set is L2-resident (x_in/x_out/acc/CSR ~148MB < 192MB L2).
// =====================================================================
__global__ void k_spmm_csr(const unsigned* __restrict__ rowptr,
                           const int* __restrict__ csrc, const float* __restrict__ cval,
                           const float* __restrict__ xin, float* __restrict__ xout,
                           float* __restrict__ acc) {
  int t = threadIdx.x;
  int lane = t & 15;                              // float4 slot within row
  int row = blockIdx.x * 16 + (t >> 4);
  if (row >= NNODES) return;
  unsigned kb = rowptr[row];
  unsigned ke = rowptr[row + 1];
  const float4* xin4 = (const float4*)xin;
  float4 s; s.x = s.y = s.z = s.w = 0.f;
  for (unsigned k = kb; k < ke; ++k) {
    int   src = csrc[k];                          // uniform across 16-lane group
    float v   = cval[k];
    float4 xv = xin4[(size_t)src * V4ROW + lane]; // global_load_b128, 256B/edge coalesced
    s.x = fmaf(v, xv.x, s.x);
    s.y = fmaf(v, xv.y, s.y);
    s.z = fmaf(v, xv.z, s.z);
    s.w = fmaf(v, xv.w, s.w);
  }
  size_t o = (size_t)row * V4ROW + lane;
  ((float4*)xout)[o] = s;
  float4* a4 = (float4*)acc;
  float4 av = a4[o];
  av.x += s.x; av.y += s.y; av.z += s.z; av.w += s.w;
  a4[o] = av;
}

// =====================================================================
// Fallback SpMM (only if workspace too small for CSR): edge scatter with
// fp32 atomics. 16 lanes per edge, 4 atomics each.
// =====================================================================
__global__ void k_spmm_atomic(const int* __restrict__ esrc, const int* __restrict__ edst,
                              const float* __restrict__ eval,
                              const float* __restrict__ xin, float* __restrict__ xout) {
  int t = blockIdx.x * blockDim.x + threadIdx.x;
  int e = t >> 4;
  int lane = t & 15;
  if (e >= NNZ_E) return;
  int   src = esrc[e];
  int   dst = edst[e];
  float v   = eval[e];
  float4 xv = ((const float4*)xin)[(size_t)src * V4ROW + lane];
  float* o = xout + (size_t)dst * DIM_C + lane * 4;
  atomicAdd(o + 0, v * xv.x);
  atomicAdd(o + 1, v * xv.y);
  atomicAdd(o + 2, v * xv.z);
  atomicAdd(o + 3, v * xv.w);
}

__global__ void k_accadd(float4* __restrict__ acc, const float4* __restrict__ x, int n4) {
  int i = blockIdx.x * blockDim.x + threadIdx.x;
  if (i >= n4) return;
  float4 a = acc[i], b = x[i];
  a.x += b.x; a.y += b.y; a.z += b.z; a.w += b.w;
  acc[i] = a;
}

// =====================================================================
// scores: diag(U . I^T) via V_WMMA_F32_16X16X4_F32, 16 pairs per wave.
// A (16x4 f32): lane L (m=L&15) holds K = {2*(L>>4), 2*(L>>4)+1} of chunk.
// B (4x16 f32): symmetric layout, lane L supplies column n=L&15.
// D (16x16 f32): lane L, reg r = D[r + 8*(L>>4)][L&15]; diagonal owned by
// lane with r = n - 8*(L>>4) in [0,8).
// light_out = acc/4 on both sides  ->  score = dot(acc_u, acc_i) / 16.
// =====================================================================
__global__ void k_scores(const int* __restrict__ users, const int* __restrict__ items,
                         const float* __restrict__ acc, float* __restrict__ out) {
  int lane = threadIdx.x & 31;
  int wid  = (blockIdx.x * blockDim.x + threadIdx.x) >> 5;  // 0..255 tiles
  int p  = lane & 15;                                       // pair within tile
  int hi = lane >> 4;
  int b  = wid * 16 + p;                                    // 0..4095 (all valid)
  const float* U = acc + (size_t)users[b] * DIM_C;
  const float* I = acc + (size_t)(N_USERS_C + items[b]) * DIM_C;
  int r = p - 8 * hi;                                       // diag register if in [0,8)
#if defined(__HIP_DEVICE_COMPILE__) && __has_builtin(__builtin_amdgcn_wmma_f32_16x16x4_f32)
  v8f c = {0.f, 0.f, 0.f, 0.f, 0.f, 0.f, 0.f, 0.f};
  #pragma unroll
  for (int k0 = 0; k0 < DIM_C; k0 += 4) {
    int ko = k0 + 2 * hi;
    v2f a;  a.x  = U[ko]; a.y  = U[ko + 1];
    v2f bb; bb.x = I[ko]; bb.y = I[ko + 1];
    c = __builtin_amdgcn_wmma_f32_16x16x4_f32(
        /*neg_a=*/false, a, /*neg_b=*/false, bb,
        /*c_mod=*/(short)0, c, /*reuse_a=*/false, /*reuse_b=*/false);
  }
  if (r >= 0 && r < 8) {
    float d = c[r] * (1.0f / 16.0f);
    out[b] = 1.0f / (1.0f + expf(-d));
  }
#else
  if (r >= 0 && r < 8) {
    float d = 0.f;
    #pragma unroll 8
    for (int k = 0; k < DIM_C; ++k) d = fmaf(U[k], I[k], d);
    d *= (1.0f / 16.0f);
    out[b] = 1.0f / (1.0f + expf(-d));
  }
#endif
}

// =====================================================================
// host launcher
// =====================================================================
extern "C" void kernel_launch(void* const* d_in, const int* in_sizes, int n_in,
                              void* d_out, int out_size, void* d_ws, size_t ws_size,
                              hipStream_t stream) {
  (void)in_sizes; (void)n_in; (void)out_size;
  const int*   users = (const int*)d_in[0];
  const int*   items = (const int*)d_in[1];
  const int*   esrc  = (const int*)d_in[2];
  const int*   edst  = (const int*)d_in[3];
  const float* eval  = (const float*)d_in[4];
  const float* uemb  = (const float*)d_in[5];
  const float* iemb  = (const float*)d_in[6];
  float* out = (float*)d_out;

  // ---- workspace carve-up (256B aligned regions) ----
  char* w = (char*)d_ws;
  size_t off = 0;
  auto carve = [&](size_t bytes) -> size_t {
    size_t o = off;
    off += (bytes + 255) & ~(size_t)255;
    return o;
  };
  const size_t xbytes = (size_t)NNODES * DIM_C * sizeof(float);   // 38.4 MB
  float* x0  = (float*)(w + carve(xbytes));
  float* x1  = (float*)(w + carve(xbytes));
  float* acc = (float*)(w + carve(xbytes));
  unsigned* counts = (unsigned*)(w + carve((size_t)NNODES * 4));
  unsigned* rowptr = (unsigned*)(w + carve((size_t)(NNODES + 1) * 4));
  unsigned* cursor = (unsigned*)(w + carve((size_t)NNODES * 4));
  unsigned* bsums  = (unsigned*)(w + carve(1024 * 4));
  int*   csrc = (int*)(w + carve((size_t)NNZ_E * 4));
  float* cval = (float*)(w + carve((size_t)NNZ_E * 4));
  const size_t need_csr = off;
  const bool use_csr = (ws_size >= need_csr);

  const int total4 = NNODES * V4ROW;                 // 2.4M float4s
  const int b_init = (total4 + 255) / 256;
  const int b_edge = (NNZ_E + 255) / 256;
  const int b_node = (NNODES + 255) / 256;           // 586 (scan blocks)
  const int b_rows = (NNODES + 15) / 16;             // 16 rows per 256-thread block

  // x_cur = acc = all_emb
  k_init<<<b_init, 256, 0, stream>>>(uemb, iemb, x0, acc);

  float* xi = x0;
  float* xo = x1;

  if (use_csr) {
    // build CSR (dst-major) once per call
    k_zero_u32<<<b_node, 256, 0, stream>>>(counts, NNODES);
    k_hist <<<b_edge, 256, 0, stream>>>(edst, counts);
    k_scan1<<<b_node, 256, 0, stream>>>(counts, rowptr, bsums, NNODES);
    k_scan2<<<1, 1024, 0, stream>>>(bsums, b_node);
    k_scan3<<<b_node, 256, 0, stream>>>(rowptr, bsums, cursor, NNODES);
    k_fill <<<b_edge, 256, 0, stream>>>(esrc, edst, eval, cursor, csrc, cval);

    for (int l = 0; l < NLAYERS; ++l) {
      k_spmm_csr<<<b_rows, 256, 0, stream>>>(rowptr, csrc, cval, xi, xo, acc);
      float* t = xi; xi = xo; xo = t;
    }
  } else {
    // atomic fallback (workspace too small for CSR arrays)
    const int b_scat = (NNZ_E * 16 + 255) / 256;
    for (int l = 0; l < NLAYERS; ++l) {
      k_zero_f4<<<b_init, 256, 0, stream>>>((float4*)xo, total4);
      k_spmm_atomic<<<b_scat, 256, 0, stream>>>(esrc, edst, eval, xi, xo);
      k_accadd<<<b_init, 256, 0, stream>>>((float4*)acc, (const float4*)xo, total4);
      float* t = xi; xi = xo; xo = t;
    }
  }

  // 4096 pairs, 16 per wave, 8 waves per block -> 32 blocks
  k_scores<<<(BATCH_C / 16) * 32 / 256, 256, 0, stream>>>(users, items, acc, out);
}